// StructureMask_72885595013728
// MI455X (gfx1250) — compile-verified
//
#include <hip/hip_runtime.h>

// ---------------------------------------------------------------------------
// StructureMask on MI455X (gfx1250)
//
//   Kernel 1 (WMMA): per-node fused projection table qv[N][64]:
//       qv[i][ 0..31] = x[i] @ Wq^T + bq
//       qv[i][32..63] = x[i] @ Wv^T + bv
//     V_WMMA_F32_16X16X4_F32, one wave = 16 nodes x 64 cols, K=128 step 4.
//   Kernel 2: per-edge gather of 32+32 floats from the L2-resident table,
//     chunked dot, sigmoid, scale, write [4, E].
//
//   Cache policy: streamed-once traffic (x, edge indices, output) uses
//   non-temporal hints so the 25.6 MB qv table keeps L2 residency.
// ---------------------------------------------------------------------------

typedef __attribute__((ext_vector_type(2))) float v2f;
typedef __attribute__((ext_vector_type(8))) float v8f;

#define HID 128
#define QVW 64   // fused q|v row width

__global__ __launch_bounds__(256) void sm_proj_wmma(
    const float* __restrict__ x,
    const float* __restrict__ Wq, const float* __restrict__ bq,
    const float* __restrict__ Wv, const float* __restrict__ bv,
    float* __restrict__ qv, int n_nodes)
{
    const int lane  = threadIdx.x & 31;
    const int wave  = threadIdx.x >> 5;
    const int slice = blockIdx.x * 8 + wave;   // 16-node slice, wave-uniform
    const int row0  = slice * 16;
    if (row0 >= n_nodes) return;               // uniform per wave: EXEC stays full

    const int lrow  = lane & 15;               // M (A) / N (B,C,D) index in tile
    const int khalf = (lane >> 4) * 2;         // lanes 16-31 hold K+2,K+3
    const float* xrow = x + (size_t)(row0 + lrow) * HID;
    const float* wq0  = Wq + (size_t)lrow * HID;
    const float* wq1  = Wq + (size_t)(lrow + 16) * HID;
    const float* wv0  = Wv + (size_t)lrow * HID;
    const float* wv1  = Wv + (size_t)(lrow + 16) * HID;

    v8f acc0 = {}, acc1 = {}, acc2 = {}, acc3 = {};

    #pragma unroll 8
    for (int k0 = 0; k0 < HID; k0 += 4) {
        const int ko = k0 + khalf;
        // x rows are touched exactly once -> non-temporal load
        v2f a  = __builtin_nontemporal_load((const v2f*)(xrow + ko));
        // weights are re-read by every wave -> regular temporal loads
        v2f b0 = *(const v2f*)(wq0 + ko);
        v2f b1 = *(const v2f*)(wq1 + ko);
        v2f b2 = *(const v2f*)(wv0 + ko);
        v2f b3 = *(const v2f*)(wv1 + ko);
        // D = A(16x4 f32) * B(4x16 f32) + C   -> v_wmma_f32_16x16x4_f32
        acc0 = __builtin_amdgcn_wmma_f32_16x16x4_f32(false, a, false, b0,
                                                     (short)0, acc0, false, false);
        acc1 = __builtin_amdgcn_wmma_f32_16x16x4_f32(false, a, false, b1,
                                                     (short)0, acc1, false, false);
        acc2 = __builtin_amdgcn_wmma_f32_16x16x4_f32(false, a, false, b2,
                                                     (short)0, acc2, false, false);
        acc3 = __builtin_amdgcn_wmma_f32_16x16x4_f32(false, a, false, b3,
                                                     (short)0, acc3, false, false);
    }

    // bias depends only on the output column (lane for C/D layout)
    const float biasq0 = bq[lrow];
    const float biasq1 = bq[lrow + 16];
    const float biasv0 = bv[lrow];
    const float biasv1 = bv[lrow + 16];

    // C/D layout: VGPR r holds M = r (lanes 0-15) or M = r+8 (lanes 16-31)
    const int mhi = (lane >> 4) * 8;
    #pragma unroll
    for (int r = 0; r < 8; ++r) {
        const int node = row0 + mhi + r;
        float* orow = qv + (size_t)node * QVW;
        // regular stores: we WANT qv resident in L2 for the edge pass
        orow[lrow]      = acc0[r] + biasq0;
        orow[lrow + 16] = acc1[r] + biasq1;
        orow[lrow + 32] = acc2[r] + biasv0;
        orow[lrow + 48] = acc3[r] + biasv1;
    }
}

__global__ __launch_bounds__(256) void sm_edge(
    const int* __restrict__ e,          // [2, E] int32
    const float* __restrict__ qv,       // [N, 64] fused q|v table (L2-resident)
    const float* __restrict__ ratio_p,
    float* __restrict__ out,            // [4, E]
    int nE)
{
    const int i = blockIdx.x * blockDim.x + threadIdx.x;
    if (i >= nE) return;

    // indices are streamed exactly once -> non-temporal
    const size_t e0 = (size_t)(unsigned)__builtin_nontemporal_load(e + i);
    const size_t e1 = (size_t)(unsigned)__builtin_nontemporal_load(e + nE + i);
    const float4* q = (const float4*)(qv + e0 * QVW);        // cols 0..31
    const float4* v = (const float4*)(qv + e1 * QVW + 32);   // cols 32..63

    const float scale = (1.0f - ratio_p[0]) * 0.5f;

    float s[4];
    #pragma unroll
    for (int c = 0; c < 4; ++c) {   // 4 chunks of 8 = 2x float4
        float4 q0 = q[2 * c], q1 = q[2 * c + 1];   // RT: keep qv in L2
        float4 v0 = v[2 * c], v1 = v[2 * c + 1];
        s[c] = q0.x * v0.x + q0.y * v0.y + q0.z * v0.z + q0.w * v0.w
             + q1.x * v1.x + q1.y * v1.y + q1.z * v1.z + q1.w * v1.w;
    }

    #pragma unroll
    for (int c = 0; c < 4; ++c) {
        const float sig = 1.0f / (1.0f + __expf(-s[c]));
        // output written once, never re-read -> non-temporal store
        __builtin_nontemporal_store(sig * scale + 1.0f,
                                    out + (size_t)c * (size_t)nE + i);
    }
}

extern "C" void kernel_launch(void* const* d_in, const int* in_sizes, int n_in,
                              void* d_out, int out_size, void* d_ws, size_t ws_size,
                              hipStream_t stream) {
    const float* x     = (const float*)d_in[0];   // [N, 128]
    const int*   e     = (const int*)d_in[1];     // [2, E]
    const float* ratio = (const float*)d_in[2];   // scalar
    const float* Wq    = (const float*)d_in[3];   // [32, 128]
    const float* bq    = (const float*)d_in[4];   // [32]
    const float* Wv    = (const float*)d_in[5];   // [32, 128]
    const float* bv    = (const float*)d_in[6];   // [32]
    float* out = (float*)d_out;                   // [4, E]

    const int n_nodes = in_sizes[0] / HID;        // 100000
    const int nE      = in_sizes[1] / 2;          // 1600000

    float* qv = (float*)d_ws;                     // [n_nodes, 64] = 25.6 MB

    // Kernel 1: fused per-node projections via WMMA f32 16x16x4.
    // 8 waves/block, each wave owns a 16-node slice -> 128 nodes per block.
    const int slices = (n_nodes + 15) / 16;
    const int blocks1 = (slices + 7) / 8;
    sm_proj_wmma<<<blocks1, 256, 0, stream>>>(x, Wq, bq, Wv, bv, qv, n_nodes);

    // Kernel 2: edge gather + chunk-dot + sigmoid (same stream -> ordered).
    const int blocks2 = (nE + 255) / 256;
    sm_edge<<<blocks2, 256, 0, stream>>>(e, qv, ratio, out, nE);
}